// Probabilistic_DAG_Generator_From_Roots_1374389535095
// MI455X (gfx1250) — compile-verified
//
#include <hip/hip_runtime.h>
#include <stdint.h>

#define N   512
#define NW  16          // 512 bits / 32 = words per bitmask row
#define EPSF 1e-10f

// Straight-through binary Gumbel-softmax, hard forward value:
// hard = (log(p+eps)+g0) > (log(1-p+eps)+g1), p = sigmoid(x)
__device__ __forceinline__ bool st_hard(float x, float u0, float u1) {
    float p  = 1.0f / (1.0f + expf(-x));
    float g0 = -logf(-logf(u0 + EPSF) + EPSF);
    float g1 = -logf(-logf(u1 + EPSF) + EPSF);
    float z0 = logf(p + EPSF) + g0;
    float z1 = logf(1.0f - p + EPSF) + g1;
    return z0 > z1;
}

// Kernel 1: fully parallel sampling. Packs edge samples into a bitmask
// (wave32 ballot -> one u32 per 32 columns), samples roots, zeroes dag.
__global__ __launch_bounds__(256) void sample_kernel(
    const float* __restrict__ root_probs,
    const float* __restrict__ edge_probs,
    const float* __restrict__ u_roots,
    const float* __restrict__ u_edges,
    float*       __restrict__ dag_out,   // [N*N], zero-initialized here
    uint32_t*    __restrict__ ebits,     // [N*NW] packed edge samples
    uint32_t*    __restrict__ rootbit)   // [N] 0/1 root samples
{
    int tid = blockIdx.x * blockDim.x + threadIdx.x;   // 0 .. N*N-1
    dag_out[tid] = 0.0f;

    bool eh = st_hard(edge_probs[tid], u_edges[tid], u_edges[N * N + tid]);
    uint32_t m = __builtin_amdgcn_ballot_w32(eh);      // wave32: 32 consecutive j
    if ((threadIdx.x & 31) == 0) ebits[tid >> 5] = m;  // word (i*16 + j/32)

    if (tid < N) {
        bool rh = st_hard(root_probs[tid], u_roots[tid], u_roots[N + tid]);
        rootbit[tid] = rh ? 1u : 0u;
    }
}

// Kernel 2: single-workgroup sequential BFS. 512 threads = 16 wave32.
// anc (32KB) + ebits (32KB) both live in LDS (needs CDNA5's 320KB WGP LDS).
// ebits is staged global->LDS via the gfx1250 async DMA path (ASYNCcnt).
__global__ __launch_bounds__(N) void bfs_kernel(
    const uint32_t* __restrict__ ebits_g,
    const uint32_t* __restrict__ rootbit,
    float*          __restrict__ dag)
{
    __shared__ __align__(16) uint32_t anc[N][NW];    // ancestor bitmask rows
    __shared__ __align__(16) uint32_t ebits[N][NW];  // edge sample bitmask rows
    __shared__ uint32_t wmask[NW];                   // per-wave ballot scratch
    __shared__ uint32_t inq[NW];                     // in_queue bitmask
    __shared__ int      queue[N];
    __shared__ int      sh_head, sh_tail, sh_i, sh_act;

    const int t = threadIdx.x;       // column owned by this thread
    const int w = t >> 5;            // wave id == bitmask word id
    const int l = t & 31;            // lane id == bit id
    const uint32_t ltmask = (l == 0) ? 0u : (0xFFFFFFFFu >> (32 - l));

    // ---- async DMA: stage 32KB edge bitmask into LDS (gfx1250 ASYNC path) ----
    {
        uint32_t lds_base = (uint32_t)(uintptr_t)&ebits[0][0];
        const char* gp = (const char*)ebits_g;
        #pragma unroll
        for (int rep = 0; rep < 4; ++rep) {
            uint32_t off   = (uint32_t)(rep * N + t) * 16u;  // 16B per thread per rep
            uint32_t laddr = lds_base + off;
            const char* ga = gp + off;
            asm volatile("global_load_async_to_lds_b128 %0, %1, off"
                         :: "v"(laddr), "v"(ga) : "memory");
        }
    }

    const bool my_root = rootbit[t] != 0u;

    // ancestors = identity (node is its own ancestor)
    #pragma unroll
    for (int q = 0; q < NW; ++q) anc[t][q] = (q == w) ? (1u << l) : 0u;

    uint32_t rmask = __builtin_amdgcn_ballot_w32(my_root);
    if (l == 0) { wmask[w] = rmask; inq[w] = rmask; }
    if (t == 0) sh_head = 0;
    __syncthreads();

    // initial queue = root indices in ascending order (ordered compaction)
    {
        int off = 0, tot = 0;
        #pragma unroll
        for (int q = 0; q < NW; ++q) {
            int c = __popc(wmask[q]);
            if (q < w) off += c;
            tot += c;
        }
        if (my_root) queue[off + __popc(rmask & ltmask)] = t;
        if (t == 0) sh_tail = tot;
    }
    asm volatile("s_wait_asynccnt 0" ::: "memory");  // ebits landed in LDS
    __syncthreads();

    // ---- sequential BFS: n steps, one queue pop per step ----
    for (int it = 0; it < N; ++it) {
        if (t == 0) {
            int act = (sh_head < sh_tail) ? 1 : 0;
            sh_act  = act;
            sh_i    = act ? queue[sh_head] : 0;
            sh_head += act;
        }
        __syncthreads();
        const int i   = sh_i;
        const int act = sh_act;

        // row i: children candidates = edge & !ancestor(i) & !root
        bool anc_ij = (anc[i][w] >> l) & 1u;
        bool eb     = (ebits[i][w] >> l) & 1u;
        bool rowb   = act && eb && !anc_ij && !my_root;
        if (act) dag[i * N + t] = rowb ? 1.0f : 0.0f;

        // ancestor propagation: child row |= row i (row i itself never a child)
        if (rowb) {
            #pragma unroll
            for (int q = 0; q < NW; ++q) anc[t][q] |= anc[i][q];
        }

        bool newb = rowb && !((inq[w] >> l) & 1u);   // newly discovered
        uint32_t nm = __builtin_amdgcn_ballot_w32(newb);
        if (l == 0) wmask[w] = nm;
        __syncthreads();

        // ordered append (ascending index) + dedup bookkeeping
        int tl = sh_tail;
        int off = 0, tot = 0;
        #pragma unroll
        for (int q = 0; q < NW; ++q) {
            int c = __popc(wmask[q]);
            if (q < w) off += c;
            tot += c;
        }
        if (newb) queue[tl + off + __popc(nm & ltmask)] = t;
        if (l == 0) inq[w] |= wmask[w];
        __syncthreads();
        if (t == 0) sh_tail = tl + tot;
    }
}

extern "C" void kernel_launch(void* const* d_in, const int* in_sizes, int n_in,
                              void* d_out, int out_size, void* d_ws, size_t ws_size,
                              hipStream_t stream) {
    (void)in_sizes; (void)n_in; (void)out_size; (void)ws_size;
    const float* root_probs = (const float*)d_in[0];
    const float* edge_probs = (const float*)d_in[1];
    const float* u_roots    = (const float*)d_in[2];
    const float* u_edges    = (const float*)d_in[3];
    float* dag = (float*)d_out;

    uint32_t* ebits   = (uint32_t*)d_ws;                                   // 32 KB
    uint32_t* rootbit = (uint32_t*)((char*)d_ws + (size_t)N * NW * 4);     // 2 KB

    sample_kernel<<<(N * N) / 256, 256, 0, stream>>>(
        root_probs, edge_probs, u_roots, u_edges, dag, ebits, rootbit);
    bfs_kernel<<<1, N, 0, stream>>>(ebits, rootbit, dag);
}